// GraphConsis_43379169689675
// MI455X (gfx1250) — compile-verified
//
#include <hip/hip_runtime.h>
#include <hip/hip_bf16.h>
#include <math.h>

// Problem constants (match reference)
#define BB   4096
#define HH   50
#define AA   32
#define DD   64
#define LL   82            // H + A
#define K2D  128           // 2*D

typedef __attribute__((ext_vector_type(2))) float v2f;
typedef __attribute__((ext_vector_type(8))) float v8f;

// One side of the encoder: gathers + softmax-attention + lin1 (WMMA) + ReLU.
// Block = 128 threads (4 waves, wave32). Each block produces 16 rows of out[B,64].
__global__ __launch_bounds__(128)
void encode_kernel(const float* __restrict__ node_tab,   // [N, 64]
                   const float* __restrict__ hist_tab,   // [N, 64]
                   const float* __restrict__ neigh_tab,  // [N, 64]
                   const float* __restrict__ r2e,        // [7, 64]
                   const float* __restrict__ att,        // [128]
                   const float* __restrict__ lin1_W,     // [64, 128] row-major
                   const float* __restrict__ lin1_b,     // [64]
                   const int*   __restrict__ nodes,      // [B]
                   const int*   __restrict__ hn,         // [B, 50]
                   const int*   __restrict__ hr,         // [B, 50]
                   const int*   __restrict__ adj,        // [B, 32]
                   float*       __restrict__ out)        // [B, 64]
{
    __shared__ float comb[16][K2D];     // [row][k] : nf(0:64) | neigh(64:128)
    __shared__ int   idx_s[4][LL];      // per-wave gathered row indices
    __shared__ float p_s[4][LL];        // per-wave softmax weights
    __shared__ float rate_logit[8];     // r2e[r] . att[64:128]
    __shared__ float att_lo[DD];        // att[0:64]

    const int tid  = threadIdx.x;
    const int wave = tid >> 5;
    const int lane = tid & 31;
    const int row0 = blockIdx.x * 16;

    if (tid < DD) att_lo[tid] = att[tid];
    if (tid < 7) {
        float s = 0.f;
        #pragma unroll
        for (int k = 0; k < DD; ++k) s += r2e[tid * DD + k] * att[DD + k];
        rate_logit[tid] = s;
    }
    __syncthreads();

    // ---------------- Phase 1: attention per row (4 rows per wave) -----------
    for (int rr = 0; rr < 4; ++rr) {
        const int row = wave * 4 + rr;
        const int b   = row0 + row;

        // Each lane owns entries e = lane, lane+32, lane+64 (e < 82).
        float lg[3];
        float m = -1e30f;
        #pragma unroll
        for (int g = 0; g < 3; ++g) {
            const int e = g * 32 + lane;
            if (e < LL) {
                int idx, r;
                const float* tab;
                if (e < HH) { idx = hn[b * HH + e];       r = hr[b * HH + e]; tab = hist_tab; }
                else        { idx = adj[b * AA + (e-HH)]; r = 6;              tab = neigh_tab; }
                idx_s[wave][e] = idx;
                const float* ep = tab + (size_t)idx * DD;
                float s = 0.f;
                #pragma unroll
                for (int k = 0; k < DD; ++k) s += ep[k] * att_lo[k];
                lg[g] = s + rate_logit[r];
                m = fmaxf(m, lg[g]);
            } else {
                lg[g] = -1e30f;
            }
        }
        // wave32 max-reduce
        #pragma unroll
        for (int off = 16; off > 0; off >>= 1)
            m = fmaxf(m, __shfl_xor(m, off, 32));
        // exp + sum-reduce
        float ex[3];
        float ssum = 0.f;
        #pragma unroll
        for (int g = 0; g < 3; ++g) {
            const int e = g * 32 + lane;
            ex[g] = (e < LL) ? __expf(lg[g] - m) : 0.f;
            ssum += ex[g];
        }
        #pragma unroll
        for (int off = 16; off > 0; off >>= 1)
            ssum += __shfl_xor(ssum, off, 32);
        const float inv = 1.f / ssum;
        #pragma unroll
        for (int g = 0; g < 3; ++g) {
            const int e = g * 32 + lane;
            if (e < LL) p_s[wave][e] = ex[g] * inv;
        }
        __syncthreads();   // uniform across block (all waves run 4 iterations)

        // Weighted sum: lane owns dims {lane, lane+32}; coalesced 128B reads.
        float acc0 = 0.f, acc1 = 0.f;
        for (int e = 0; e < LL; ++e) {
            const float* tab = (e < HH) ? hist_tab : neigh_tab;
            const float* ep  = tab + (size_t)idx_s[wave][e] * DD;
            const float  w   = p_s[wave][e];
            acc0 += w * ep[lane];
            acc1 += w * ep[lane + 32];
        }
        const size_t nbase = (size_t)nodes[b] * DD;
        comb[row][lane]      = node_tab[nbase + lane];
        comb[row][lane + 32] = node_tab[nbase + lane + 32];
        comb[row][DD + lane]      = acc0;
        comb[row][DD + lane + 32] = acc1;
    }
    __syncthreads();

    // ---------------- Phase 2: lin1 via FP32 WMMA 16x16x4 --------------------
    // out[m, n] = relu( sum_k comb[m][k] * lin1_W[n][k] + lin1_b[n] )
    // Wave `wave` owns output columns n0..n0+15. 32 k-steps of 4.
    const int n0   = wave * 16;
    const int half = lane >> 4;          // 0: K pair {0,1}; 1: K pair {2,3}
    const int lm   = lane & 15;          // A: row m; B: col n; D: col n
    v8f acc = {};
    for (int kk = 0; kk < K2D; kk += 4) {
        const int ka = kk + half * 2;
        v2f a, bf;
        a.x = comb[lm][ka];
        a.y = comb[lm][ka + 1];
        const float* wp = lin1_W + (size_t)(n0 + lm) * K2D + ka;  // B[k][n] = lin1_W[n][k]
        bf.x = wp[0];
        bf.y = wp[1];
        acc = __builtin_amdgcn_wmma_f32_16x16x4_f32(
            /*neg_a=*/false, a, /*neg_b=*/false, bf,
            /*c_mod=*/(short)0, acc, /*reuse_a=*/false, /*reuse_b=*/false);
    }
    // D layout: VGPR j -> (lanes 0-15: M=j, N=lane) (lanes 16-31: M=j+8, N=lane-16)
    const int   ncol = n0 + lm;
    const float bias = lin1_b[ncol];
    #pragma unroll
    for (int j = 0; j < 8; ++j) {
        const int mrow = j + half * 8;
        const float v = acc[j] + bias;
        out[(size_t)(row0 + mrow) * DD + ncol] = fmaxf(v, 0.f);
    }
}

// Final rowwise dot: out[b] = sum_d eu[b,d] * ev[b,d]
__global__ __launch_bounds__(256)
void dot_kernel(const float* __restrict__ eu, const float* __restrict__ ev,
                float* __restrict__ out, int n)
{
    const int row  = blockIdx.x * 8 + (threadIdx.x >> 5);
    const int lane = threadIdx.x & 31;
    if (row >= n) return;
    const size_t base = (size_t)row * DD;
    float s = eu[base + lane] * ev[base + lane]
            + eu[base + lane + 32] * ev[base + lane + 32];
    #pragma unroll
    for (int off = 16; off > 0; off >>= 1)
        s += __shfl_xor(s, off, 32);
    if (lane == 0) out[row] = s;
}

extern "C" void kernel_launch(void* const* d_in, const int* in_sizes, int n_in,
                              void* d_out, int out_size, void* d_ws, size_t ws_size,
                              hipStream_t stream) {
    const float* u2e    = (const float*)d_in[0];
    const float* v2e    = (const float*)d_in[1];
    const float* r2e    = (const float*)d_in[2];
    const float* att    = (const float*)d_in[3];
    // d_in[4] agg_W, d_in[5] agg_b: dead code in the reference (unused _q)
    const float* lin1_W = (const float*)d_in[6];
    const float* lin1_b = (const float*)d_in[7];
    const int* nodes_u  = (const int*)d_in[8];
    const int* nodes_v  = (const int*)d_in[9];
    const int* hist_u   = (const int*)d_in[10];
    const int* hist_ur  = (const int*)d_in[11];
    const int* adj_u    = (const int*)d_in[12];
    const int* hist_v   = (const int*)d_in[13];
    const int* hist_vr  = (const int*)d_in[14];
    const int* adj_v    = (const int*)d_in[15];

    float* eu = (float*)d_ws;
    float* ev = eu + (size_t)BB * DD;
    float* outp = (float*)d_out;

    const int blocks = BB / 16;
    // user side: node=u2e, hist=v2e, neighbor=u2e
    encode_kernel<<<blocks, 128, 0, stream>>>(u2e, v2e, u2e, r2e, att, lin1_W, lin1_b,
                                              nodes_u, hist_u, hist_ur, adj_u, eu);
    // item side: node=v2e, hist=u2e, neighbor=v2e
    encode_kernel<<<blocks, 128, 0, stream>>>(v2e, u2e, v2e, r2e, att, lin1_W, lin1_b,
                                              nodes_v, hist_v, hist_vr, adj_v, ev);
    dot_kernel<<<(BB + 7) / 8, 256, 0, stream>>>(eu, ev, outp, BB);
}